// UnifiedAutoencoderTransformer_65481071409439
// MI455X (gfx1250) — compile-verified
//
#include <hip/hip_runtime.h>
#include <hip/hip_bf16.h>
#include <math.h>

typedef __attribute__((ext_vector_type(2))) float v2f;
typedef __attribute__((ext_vector_type(8))) float v8f;

#define BQ     4096
#define DIN    1024
#define DD     8
#define LL     2
#define FFN    128
#define NQ     10
#define DIMQ   1024      // 2^10
#define REPS   5
#define ROWS_PB 4        // statevector rows per block (32 KB LDS)

// ---------------------------------------------------------------------------
// Kernel 1: encoder GEMM  z_pre = x @ enc_W + enc_b   (4096x1024)@(1024x8)
// 128 threads = 4 waves; block covers 32 rows (two 16-row M-tiles).
// Each wave: one K-quarter (256 K values, 64 k-steps), TWO accumulators
// sharing the same B registers -> 2 independent WMMA chains per wave and
// half the B traffic. Partials reduced through LDS.
// B padded to N=16: cols 8..15 masked to zero via clamped in-bounds address
// + 0/1 multiply (keeps loads unconditional, no exec branches).
// ---------------------------------------------------------------------------
__global__ __launch_bounds__(128) void enc_gemm_wmma(
    const float* __restrict__ x, const float* __restrict__ encW,
    const float* __restrict__ encB, float* __restrict__ zpre)
{
    __shared__ float red[4][2][32][8];     // [kwave][mtile][lane][j]

    const int lane = threadIdx.x & 31;
    const int wave = threadIdx.x >> 5;
    const int l16  = lane & 15;
    const int half = lane >> 4;            // 0: K,K+1   1: K+2,K+3
    const int row0 = blockIdx.x * 32;

    const int   bcol  = l16 & 7;           // clamped, always in-bounds
    const float bmask = (l16 < DD) ? 1.0f : 0.0f;

    const float* xrow0 = x + (size_t)(row0 + l16) * DIN;
    const float* xrow1 = xrow0 + (size_t)16 * DIN;
    const int k0 = wave * (DIN / 4);       // 256 K values per wave

    v8f c0 = {}, c1 = {};
    for (int k = k0; k < k0 + DIN / 4; k += 4) {
        const int ka = k + half * 2;
        // prefetch the A streams ~1KB ahead (global_prefetch_b8)
        __builtin_prefetch(xrow0 + ka + 256, 0, 3);
        __builtin_prefetch(xrow1 + ka + 256, 0, 3);
        v2f a0, a1, b;
        a0.x = xrow0[ka];
        a0.y = xrow0[ka + 1];
        a1.x = xrow1[ka];
        a1.y = xrow1[ka + 1];
        b.x = bmask * encW[ka * DD + bcol];
        b.y = bmask * encW[(ka + 1) * DD + bcol];
        c0 = __builtin_amdgcn_wmma_f32_16x16x4_f32(
                 false, a0, false, b, (short)0, c0, false, false);
        c1 = __builtin_amdgcn_wmma_f32_16x16x4_f32(
                 false, a1, false, b, (short)0, c1, false, false);
    }

#pragma unroll
    for (int j = 0; j < 8; ++j) {
        red[wave][0][lane][j] = c0[j];
        red[wave][1][lane][j] = c1[j];
    }
    __syncthreads();

    // waves 0 and 1 each write one M-tile's result
    if (wave < 2 && l16 < DD) {
        const int  t    = wave;
        const float bias = encB[l16];
#pragma unroll
        for (int j = 0; j < 8; ++j) {
            const float s = red[0][t][lane][j] + red[1][t][lane][j]
                          + red[2][t][lane][j] + red[3][t][lane][j];
            const int m = t * 16 + half * 8 + j;
            zpre[(size_t)(row0 + m) * DD + l16] = s + bias;
        }
    }
}

// ---------------------------------------------------------------------------
// Kernel 2: per-row transformer (D=8) + 10-qubit statevector simulation.
// 128 threads = 4 waves; each wave owns one row's 1024-amp complex state in
// LDS, so only wave-level ordering (no block barrier) is needed per gate.
// ---------------------------------------------------------------------------
__device__ inline void ln8(const float* h, const float* g, const float* b, float* out)
{
    float m = 0.f;
#pragma unroll
    for (int d = 0; d < DD; ++d) m += h[d];
    m *= 0.125f;
    float v = 0.f;
#pragma unroll
    for (int d = 0; d < DD; ++d) { const float t = h[d] - m; v += t * t; }
    v *= 0.125f;
    const float inv = 1.0f / sqrtf(v + 1e-5f);
#pragma unroll
    for (int d = 0; d < DD; ++d) out[d] = (h[d] - m) * inv * g[d] + b[d];
}

__global__ __launch_bounds__(128) void qsim_kernel(
    const float* __restrict__ zpre,
    const float* __restrict__ Wv,  const float* __restrict__ bv,
    const float* __restrict__ Wo,  const float* __restrict__ bo,
    const float* __restrict__ ln1g, const float* __restrict__ ln1b,
    const float* __restrict__ W1,  const float* __restrict__ b1,
    const float* __restrict__ W2,  const float* __restrict__ b2,
    const float* __restrict__ ln2g, const float* __restrict__ ln2b,
    const float* __restrict__ qw,  float* __restrict__ zq)
{
    __shared__ float sRe[ROWS_PB][DIMQ];
    __shared__ float sIm[ROWS_PB][DIMQ];

    const int g    = threadIdx.x >> 5;     // wave id == row group
    const int lane = threadIdx.x & 31;
    const int row  = blockIdx.x * ROWS_PB + g;

    // ---- transformer layers, computed redundantly by all 32 lanes ----
    float z[DD];
#pragma unroll
    for (int d = 0; d < DD; ++d) z[d] = zpre[(size_t)row * DD + d];

    for (int l = 0; l < LL; ++l) {
        const float* wv = Wv + l * 64;  const float* wo = Wo + l * 64;
        float v[DD], attn[DD], h[DD];
#pragma unroll
        for (int d = 0; d < DD; ++d) {
            float s = bv[l * DD + d];
#pragma unroll
            for (int k = 0; k < DD; ++k) s += z[k] * wv[k * DD + d];
            v[d] = s;
        }
#pragma unroll
        for (int d = 0; d < DD; ++d) {
            float s = bo[l * DD + d];
#pragma unroll
            for (int k = 0; k < DD; ++k) s += v[k] * wo[k * DD + d];
            attn[d] = s;
        }
#pragma unroll
        for (int d = 0; d < DD; ++d) h[d] = z[d] + attn[d];
        ln8(h, ln1g + l * DD, ln1b + l * DD, z);

        // FFN: 128 hidden units split across 32 lanes (4 each)
        float acc[DD];
#pragma unroll
        for (int d = 0; d < DD; ++d) acc[d] = 0.f;
#pragma unroll
        for (int j = 0; j < 4; ++j) {
            const int hid = lane + j * 32;
            float hv = b1[l * FFN + hid];
#pragma unroll
            for (int k = 0; k < DD; ++k) hv += z[k] * W1[l * DD * FFN + k * FFN + hid];
            hv = fmaxf(hv, 0.0f);
#pragma unroll
            for (int d = 0; d < DD; ++d) acc[d] += hv * W2[l * FFN * DD + hid * DD + d];
        }
        for (int off = 16; off > 0; off >>= 1) {
#pragma unroll
            for (int d = 0; d < DD; ++d) acc[d] += __shfl_xor(acc[d], off, 32);
        }
#pragma unroll
        for (int d = 0; d < DD; ++d) h[d] = z[d] + acc[d] + b2[l * DD + d];
        ln8(h, ln2g + l * DD, ln2b + l * DD, z);
    }

    // ---- normalize and initialize statevector in LDS ----
    float nrm2 = 0.f;
#pragma unroll
    for (int d = 0; d < DD; ++d) nrm2 += z[d] * z[d];
    const float inv = 1.0f / sqrtf(nrm2);

    for (int i = lane; i < DIMQ; i += 32) { sRe[g][i] = 0.f; sIm[g][i] = 0.f; }
#pragma unroll
    for (int d = 0; d < DD; ++d) if (lane == d) sRe[g][d] = z[d] * inv;
    __builtin_amdgcn_wave_barrier();

    // ---- circuit: REPS x (10 RX gates + CNOT ring) ----
    for (int r = 0; r < REPS; ++r) {
        for (int w = 0; w < NQ; ++w) {
            const float th  = 0.5f * qw[r * NQ + w];
            const float cth = cosf(th);
            const float sth = sinf(th);
            const int bit = 1 << (NQ - 1 - w);
            const int low = bit - 1;
            for (int p = lane; p < DIMQ / 2; p += 32) {
                const int i0 = ((p & ~low) << 1) | (p & low);
                const int i1 = i0 | bit;
                const float a0r = sRe[g][i0], a0i = sIm[g][i0];
                const float a1r = sRe[g][i1], a1i = sIm[g][i1];
                sRe[g][i0] = cth * a0r + sth * a1i;
                sIm[g][i0] = cth * a0i - sth * a1r;
                sRe[g][i1] = cth * a1r + sth * a0i;
                sIm[g][i1] = cth * a1i - sth * a0r;
            }
            __builtin_amdgcn_wave_barrier();
        }
        for (int w = 0; w < NQ; ++w) {
            const int cbit = 1 << (NQ - 1 - w);
            const int tbit = 1 << (NQ - 1 - ((w + 1) % NQ));
            for (int i = lane; i < DIMQ; i += 32) {
                if ((i & cbit) && !(i & tbit)) {
                    const int j = i | tbit;
                    const float tr = sRe[g][i], ti = sIm[g][i];
                    sRe[g][i] = sRe[g][j];  sIm[g][i] = sIm[g][j];
                    sRe[g][j] = tr;         sIm[g][j] = ti;
                }
            }
            __builtin_amdgcn_wave_barrier();
        }
    }

    // ---- Z expectations for the first 8 qubits ----
    float acc[DD];
#pragma unroll
    for (int d = 0; d < DD; ++d) acc[d] = 0.f;
    for (int i = lane; i < DIMQ; i += 32) {
        const float re = sRe[g][i], im = sIm[g][i];
        const float p = re * re + im * im;
#pragma unroll
        for (int w = 0; w < DD; ++w) {
            const int bit = 1 << (NQ - 1 - w);
            acc[w] += (i & bit) ? -p : p;
        }
    }
    for (int off = 16; off > 0; off >>= 1) {
#pragma unroll
        for (int w = 0; w < DD; ++w) acc[w] += __shfl_xor(acc[w], off, 32);
    }
#pragma unroll
    for (int w = 0; w < DD; ++w) if (lane == w) zq[(size_t)row * DD + w] = acc[w];
}

// ---------------------------------------------------------------------------
// Kernel 3: decoder GEMM  out = zq @ dec_W + dec_b   (4096x8)@(8x1024)
// One wave per 16x16 output tile; K=8 -> two f32 WMMA 16x16x4 ops.
// ---------------------------------------------------------------------------
__global__ __launch_bounds__(32) void dec_gemm_wmma(
    const float* __restrict__ zq, const float* __restrict__ decW,
    const float* __restrict__ decB, float* __restrict__ out)
{
    const int lane = threadIdx.x;
    const int l16  = lane & 15;
    const int half = lane >> 4;
    const int n0 = blockIdx.x * 16;
    const int m0 = blockIdx.y * 16;

    const float* arow = zq + (size_t)(m0 + l16) * DD;

    v8f c = {};
#pragma unroll
    for (int k = 0; k < DD; k += 4) {
        const int ka = k + half * 2;
        v2f a, b;
        a.x = arow[ka];
        a.y = arow[ka + 1];
        b.x = decW[(size_t)ka * DIN + n0 + l16];
        b.y = decW[(size_t)(ka + 1) * DIN + n0 + l16];
        c = __builtin_amdgcn_wmma_f32_16x16x4_f32(
                false, a, false, b, (short)0, c, false, false);
    }

    const float bias = decB[n0 + l16];
#pragma unroll
    for (int j = 0; j < 8; ++j) {
        const int m = half * 8 + j;
        out[(size_t)(m0 + m) * DIN + n0 + l16] = c[j] + bias;
    }
}

// ---------------------------------------------------------------------------
extern "C" void kernel_launch(void* const* d_in, const int* in_sizes, int n_in,
                              void* d_out, int out_size, void* d_ws, size_t ws_size,
                              hipStream_t stream) {
    const float* x    = (const float*)d_in[0];
    const float* encW = (const float*)d_in[1];
    const float* encB = (const float*)d_in[2];
    const float* decW = (const float*)d_in[3];
    const float* decB = (const float*)d_in[4];
    const float* Wv   = (const float*)d_in[5];
    const float* bv   = (const float*)d_in[6];
    const float* Wo   = (const float*)d_in[7];
    const float* bo   = (const float*)d_in[8];
    const float* ln1g = (const float*)d_in[9];
    const float* ln1b = (const float*)d_in[10];
    const float* W1   = (const float*)d_in[11];
    const float* b1   = (const float*)d_in[12];
    const float* W2   = (const float*)d_in[13];
    const float* b2   = (const float*)d_in[14];
    const float* ln2g = (const float*)d_in[15];
    const float* ln2b = (const float*)d_in[16];
    const float* qw   = (const float*)d_in[17];
    float* out  = (float*)d_out;

    float* zpre = (float*)d_ws;              // 4096*8 floats
    float* zqv  = zpre + (size_t)BQ * DD;    // 4096*8 floats

    enc_gemm_wmma<<<BQ / 32, 128, 0, stream>>>(x, encW, encB, zpre);

    qsim_kernel<<<BQ / ROWS_PB, 128, 0, stream>>>(
        zpre, Wv, bv, Wo, bo, ln1g, ln1b, W1, b1, W2, b2, ln2g, ln2b, qw, zqv);

    dim3 dgrid(DIN / 16, BQ / 16);
    dec_gemm_wmma<<<dgrid, 32, 0, stream>>>(zqv, decW, decB, out);
}